// AttentionCropLayer_28166395527344
// MI455X (gfx1250) — compile-verified
//
#include <hip/hip_runtime.h>
#include <stdint.h>

// RA-CNN AttentionCropLayer for MI455X (gfx1250).
// Store-bandwidth-bound crop+mask+bilinear-resize:
//  - Tensor Data Mover DMAs the strided 2D crop (n x n f32, row stride 224)
//    into packed LDS (TENSORcnt), overlapped with coefficient-table setup.
//  - One crop-local coefficient table serves rows AND columns (depends only
//    on crop size n); column coefficients are hoisted into registers so LDS
//    traffic (~20 B/px) stays under the 64-bank budget while stores run at
//    HBM rate.
//  - Coalesced global_store_b128 output.

#define IMG 224
#define HALF_ 112
#define OFFS 56
#define TL_BASE 38
#define TL_RANGE 18
#define MAX_IN 110   // 2*tl_max, tl <= 55

typedef unsigned int v4u __attribute__((ext_vector_type(4)));
typedef int          v8i __attribute__((ext_vector_type(8)));
typedef int          v4i __attribute__((ext_vector_type(4)));

__device__ __forceinline__ float sig10(float x) {
    return 1.0f / (1.0f + __expf(-10.0f * x));
}

__global__ __launch_bounds__(224)
void attn_crop_kernel(const float* __restrict__ apn,
                      const float* __restrict__ in,
                      float* __restrict__ out) {
    __shared__ float  tile[MAX_IN * MAX_IN];   // packed crop, row stride n
    __shared__ float4 ctab[IMG];               // {w, m0, m1, pack(i0,i1)}

    const int img = blockIdx.x;                // b*3 + ch
    const int b   = img / 3;
    const int tid = threadIdx.x;               // 0..223 (7 wave32s)

    const float a0 = apn[b * 3 + 0];
    const float a1 = apn[b * 3 + 1];
    const float a2 = apn[b * 3 + 2];
    const int tx = HALF_ + (int)truncf(a0 * (float)OFFS + 0.5f);
    const int ty = HALF_ + (int)truncf(a1 * (float)OFFS + 0.5f);
    const int tl = TL_BASE + (int)truncf((a2 + 1.0f) * 0.5f * (float)TL_RANGE);
    const int n  = 2 * tl;                     // crop side, 76..110
    const float nf = (float)n;

    if (tid < 32) {
        // Wave 0: issue the TDM DMA first so it overlaps the table setup.
        const float* src = in + (size_t)img * (IMG * IMG)
                              + (size_t)(tx - tl) * IMG + (ty - tl);
        uint64_t ga  = (uint64_t)(uintptr_t)src;
        uint32_t lds = (uint32_t)(uintptr_t)(&tile[0]);

        v4u g0;
        g0.x = 1u;                                      // count = 1 (valid D#)
        g0.y = lds;                                     // lds_addr (bytes)
        g0.z = (uint32_t)(ga & 0xFFFFFFFFu);            // global_addr[31:0]
        g0.w = (uint32_t)((ga >> 32) & 0x1FFFFFFu)      // global_addr[56:32]
             | (2u << 30);                              // type = 2 ("image")

        v8i g1;
        g1[0] = (int)(2u << 16);                        // data_size = 4B, no mcast
        g1[1] = (int)(((uint32_t)IMG) << 16);           // tensor_dim0 lo16
        g1[2] = (int)(((uint32_t)IMG) << 16);           // tensor_dim1 lo16 (dim0 hi=0)
        g1[3] = (int)(((uint32_t)n)   << 16);           // tile_dim0 (dim1 hi=0)
        g1[4] = (int)((uint32_t)n);                     // tile_dim1 (tile_dim2=0)
        g1[5] = IMG;                                    // tensor_dim0_stride lo32
        g1[6] = (int)((((uint32_t)(IMG * IMG)) & 0xFFFFu) << 16); // dim1_stride lo16
        g1[7] = (int)(((uint32_t)(IMG * IMG)) >> 16);             // dim1_stride hi
        v4i gz = {0, 0, 0, 0};                          // groups 2/3 unused (2D)
#if defined(__clang_major__) && (__clang_major__ >= 23)
        v8i z8 = {0, 0, 0, 0, 0, 0, 0, 0};
        __builtin_amdgcn_tensor_load_to_lds(g0, g1, gz, gz, z8, 0);
#else
        __builtin_amdgcn_tensor_load_to_lds(g0, g1, gz, gz, 0);
#endif
    }

    // All 224 threads: one coefficient-table entry each. In crop-local
    // coordinates the bilinear weights AND the sigmoid-mask values depend
    // only on n, so one table serves both rows and columns.
    {
        int j = tid;
        float src = fmaxf((j + 0.5f) * nf * (1.0f / (float)IMG) - 0.5f, 0.0f);
        float fi0 = floorf(src);
        int   i0  = (int)fi0;
        float w   = src - fi0;
        int   i1  = min(i0 + 1, n - 1);
        float m0  = sig10((float)i0) - sig10((float)(i0 - n));
        float m1  = sig10((float)i1) - sig10((float)(i1 - n));
        float4 e;
        e.x = w; e.y = m0; e.z = m1;
        e.w = __int_as_float((i0 & 0xFFFF) | (i1 << 16));
        ctab[j] = e;
    }

    if (tid < 32) {
        __builtin_amdgcn_s_wait_tensorcnt(0);
    }
    __syncthreads();

    // Each thread owns a fixed 4-column group; column coefficients live in
    // registers for the whole row loop (kills 16 B/px of LDS table traffic).
    const int cg    = tid % 56;                // column group (4 px wide)
    const int rbase = tid / 56;                // row phase 0..3
    float wc[4], mc0[4], mc1[4];
    int   c0[4], c1[4];
    #pragma unroll
    for (int k = 0; k < 4; ++k) {
        float4 ct = ctab[4 * cg + k];
        int cpk = __float_as_int(ct.w);
        wc[k]  = ct.x; mc0[k] = ct.y; mc1[k] = ct.z;
        c0[k]  = cpk & 0xFFFF;
        c1[k]  = cpk >> 16;
    }

    float4* __restrict__ outImg = (float4*)(out + (size_t)img * (IMG * IMG));
    // 224 rows / 4 phases = 56 iterations; 4 pixels (one float4) per iter.
    for (int it = 0; it < 56; ++it) {
        int r = rbase + 4 * it;

        float4 rt = ctab[r];                   // broadcast within 56-thread group
        int rpk = __float_as_int(rt.w);
        int r0  = rpk & 0xFFFF;
        int r1  = rpk >> 16;
        const float* row0 = &tile[r0 * n];
        const float* row1 = &tile[r1 * n];
        float wr = rt.x, mr0 = rt.y, mr1 = rt.z;

        float res[4];
        #pragma unroll
        for (int k = 0; k < 4; ++k) {
            float v00 = row0[c0[k]] * mc0[k];
            float v01 = row0[c1[k]] * mc1[k];
            float v10 = row1[c0[k]] * mc0[k];
            float v11 = row1[c1[k]] * mc1[k];
            float top = (v00 + wc[k] * (v01 - v00)) * mr0;
            float bot = (v10 + wc[k] * (v11 - v10)) * mr1;
            res[k] = top + wr * (bot - top);
        }
        float4 o; o.x = res[0]; o.y = res[1]; o.z = res[2]; o.w = res[3];
        outImg[r * 56 + cg] = o;
    }
}

extern "C" void kernel_launch(void* const* d_in, const int* in_sizes, int n_in,
                              void* d_out, int out_size, void* d_ws, size_t ws_size,
                              hipStream_t stream) {
    (void)n_in; (void)out_size; (void)d_ws; (void)ws_size;
    const float* apn    = (const float*)d_in[0];   // (B, 3) f32
    const float* inputs = (const float*)d_in[1];   // (B, 3, 224, 224) f32
    float*       out    = (float*)d_out;           // (B, 3, 224, 224) f32
    const int B = in_sizes[0] / 3;
    attn_crop_kernel<<<dim3(3 * B), dim3(224), 0, stream>>>(apn, inputs, out);
}